// GNN_MyGAT_83047487635731
// MI455X (gfx1250) — compile-verified
//
#include <hip/hip_runtime.h>
#include <math.h>

#define N_NODES   100000
#define N_EDGES   1600000
#define E_TOT     (N_EDGES + N_NODES)   // edges + self loops
#define HID       64
#define N_GRAPHS  64
#define NEG_SLOPE 0.2f
// K-pair-interleaved LDS layout for W: element (r,c) at Ws[(r>>1)*PAIR_STRIDE + 2c + (r&1)].
// PAIR_STRIDE = 128 + 32 pad so consecutive pairs sit on disjoint LDS bank halves.
#define PAIR_STRIDE 160

typedef __attribute__((ext_vector_type(2))) float v2f;
typedef __attribute__((ext_vector_type(8))) float v8f;

// ---- order-preserving float<->uint mapping for atomic max over floats ----
__device__ __forceinline__ unsigned f2ord(float f) {
    unsigned u = __float_as_uint(f);
    return (u & 0x80000000u) ? ~u : (u | 0x80000000u);
}
__device__ __forceinline__ float ord2f(unsigned u) {
    return (u & 0x80000000u) ? __uint_as_float(u & 0x7FFFFFFFu) : __uint_as_float(~u);
}

// ============================================================
// lin = A[N,64] @ W[64,64]  via V_WMMA_F32_16X16X4_F32
// one wave -> 16 rows x 64 cols; W staged in LDS, K-pair interleaved
// so each B fragment is one ds_load_b64 into an aligned VGPR pair.
// ============================================================
__global__ __launch_bounds__(256) void k_gemm(const float* __restrict__ A,
                                              const float* __restrict__ W,
                                              float* __restrict__ O) {
    __shared__ float Ws[32 * PAIR_STRIDE];   // 20 KB
    // cooperative staging: thread handles (K-pair p, column c)
    for (int i = threadIdx.x; i < 32 * HID; i += 256) {
        int p = i >> 6;        // K pair (rows 2p, 2p+1)
        int c = i & 63;        // column
        v2f v;
        v.x = W[(2 * p)     * HID + c];
        v.y = W[(2 * p + 1) * HID + c];
        *(v2f*)(Ws + p * PAIR_STRIDE + c * 2) = v;
    }
    __syncthreads();

    int wave = (blockIdx.x * blockDim.x + threadIdx.x) >> 5;
    int lane = threadIdx.x & 31;
    int m0   = wave * 16;
    if (m0 >= N_NODES) return;     // wave-uniform: EXEC stays all-ones for WMMA

    int half = lane >> 4;          // 0: lanes 0-15, 1: lanes 16-31
    int l16  = lane & 15;

    v8f acc0 = {}, acc1 = {}, acc2 = {}, acc3 = {};
    const float* arow = A + (m0 + l16) * HID + half * 2;

    #pragma unroll
    for (int k = 0; k < HID; k += 4) {
        // A fragment (16x4 f32): lane<16 -> K=k,k+1 ; lane>=16 -> K=k+2,k+3
        v2f a = *(const v2f*)(arow + k);
        // B fragments (4x16 f32): lanes 0-15 need K rows k..k+1 (pair k/2),
        // lanes 16-31 rows k+2..k+3 (pair k/2+1); both contiguous ds_load_b64.
        const float* wp = Ws + ((k >> 1) + half) * PAIR_STRIDE + l16 * 2;
        v2f b0 = *(const v2f*)(wp + 0);    // cols  0..15
        v2f b1 = *(const v2f*)(wp + 32);   // cols 16..31
        v2f b2 = *(const v2f*)(wp + 64);   // cols 32..47
        v2f b3 = *(const v2f*)(wp + 96);   // cols 48..63
        acc0 = __builtin_amdgcn_wmma_f32_16x16x4_f32(false, a, false, b0, (short)0, acc0, false, false);
        acc1 = __builtin_amdgcn_wmma_f32_16x16x4_f32(false, a, false, b1, (short)0, acc1, false, false);
        acc2 = __builtin_amdgcn_wmma_f32_16x16x4_f32(false, a, false, b2, (short)0, acc2, false, false);
        acc3 = __builtin_amdgcn_wmma_f32_16x16x4_f32(false, a, false, b3, (short)0, acc3, false, false);
    }

    // C/D layout: VGPR r, lanes 0-15 -> M=r, lanes 16-31 -> M=r+8, N=lane%16
    float* obase = O + (m0 + half * 8) * HID + l16;
    #pragma unroll
    for (int r = 0; r < 8; ++r) {
        obase[r * HID + 0]  = acc0[r];
        obase[r * HID + 16] = acc1[r];
        obase[r * HID + 32] = acc2[r];
        obase[r * HID + 48] = acc3[r];
    }
}

// ============================================================
// per-node attention scalars + segment-softmax state init
// aI[n] = lin[n] . att[0:64] ; aJ[n] = lin[n] . att[64:128]
// ============================================================
__global__ void k_alpha(const float* __restrict__ lin, const float* __restrict__ att,
                        float* __restrict__ aI, float* __restrict__ aJ,
                        unsigned* __restrict__ segmax, float* __restrict__ segsum) {
    int n = blockIdx.x * blockDim.x + threadIdx.x;
    if (n >= N_NODES) return;
    const float* h = lin + n * HID;
    float si = 0.f, sj = 0.f;
    #pragma unroll
    for (int c = 0; c < HID; ++c) {
        float v = h[c];
        si += v * att[c];
        sj += v * att[HID + c];
    }
    aI[n] = si;
    aJ[n] = sj;
    segmax[n] = 0x007FFFFFu;   // f2ord(-inf)
    segsum[n] = 0.f;
}

__global__ void k_zero(float* __restrict__ p, int n) {
    int i = blockIdx.x * blockDim.x + threadIdx.x;
    if (i < n) p[i] = 0.f;
}

// ============================================================
// per-edge logits (leaky_relu) + segment max (uint-ordered atomicMax)
// edges [0,N_EDGES): from edge_index; edges [N_EDGES,E_TOT): self loops
// ============================================================
__global__ void k_logit(const int* __restrict__ src, const int* __restrict__ dst,
                        const float* __restrict__ eattr,
                        const float* __restrict__ aI, const float* __restrict__ aJ,
                        const float* __restrict__ att,
                        float* __restrict__ logits, unsigned* __restrict__ segmax) {
    int e = blockIdx.x * blockDim.x + threadIdx.x;
    if (e >= E_TOT) return;
    int s, d; float ea; bool valid;
    if (e < N_EDGES) {
        s = src[e]; d = dst[e]; ea = eattr[e]; valid = (s != d);
    } else {
        s = d = e - N_EDGES; ea = 0.f; valid = true;
    }
    float lg = aI[d] + aJ[s] + ea * att[2 * HID];
    lg = (lg >= 0.f) ? lg : NEG_SLOPE * lg;
    if (!valid) {
        logits[e] = -INFINITY;
        return;
    }
    logits[e] = lg;
    atomicMax(segmax + d, f2ord(lg));
}

// ============================================================
// a[e] = exp(logit - max[dst]) ; segment sum of a
// ============================================================
__global__ void k_exp(const int* __restrict__ dst,
                      const unsigned* __restrict__ segmax,
                      float* __restrict__ ebuf, float* __restrict__ segsum) {
    int e = blockIdx.x * blockDim.x + threadIdx.x;
    if (e >= E_TOT) return;
    int d = (e < N_EDGES) ? dst[e] : (e - N_EDGES);
    float lg = ebuf[e];
    float a = 0.f;
    if (lg != -INFINITY) a = expf(lg - ord2f(segmax[d]));
    ebuf[e] = a;
    if (a != 0.f) atomicAdd(segsum + d, a);
}

// ============================================================
// weighted scatter-add: acc[dst] += (a/denom[dst]) * lin[src]
// 16 threads per edge, float4 gathers, coalesced float atomics
// ============================================================
__global__ void k_scatter(const int* __restrict__ src, const int* __restrict__ dst,
                          const float* __restrict__ ebuf, const float* __restrict__ segsum,
                          const float* __restrict__ lin, float* __restrict__ acc) {
    int tid = blockIdx.x * blockDim.x + threadIdx.x;
    int e = tid >> 4;
    if (e >= E_TOT) return;
    float a = ebuf[e];
    if (a == 0.f) return;
    int g = tid & 15;
    int s, d;
    if (e < N_EDGES) { s = src[e]; d = dst[e]; }
    else             { s = d = e - N_EDGES; }
    float coef = a / segsum[d];
    float4 v = ((const float4*)(lin + s * HID))[g];
    float* op = acc + d * HID + g * 4;
    atomicAdd(op + 0, coef * v.x);
    atomicAdd(op + 1, coef * v.y);
    atomicAdd(op + 2, coef * v.z);
    atomicAdd(op + 3, coef * v.w);
}

// h = relu(acc + bias)
__global__ void k_final(const float* __restrict__ acc, const float* __restrict__ b,
                        float* __restrict__ h) {
    int i = blockIdx.x * blockDim.x + threadIdx.x;
    if (i >= N_NODES * HID) return;
    float v = acc[i] + b[i & (HID - 1)];
    h[i] = (v > 0.f) ? v : 0.f;
}

__global__ void k_poolzero(float* __restrict__ pooled) {
    int i = blockIdx.x * blockDim.x + threadIdx.x;
    if (i < N_GRAPHS * HID) pooled[i] = 0.f;
}

// pooled[batch[n]] += h[n]
__global__ void k_pool(const float* __restrict__ h, const int* __restrict__ batch,
                       float* __restrict__ pooled) {
    int i = blockIdx.x * blockDim.x + threadIdx.x;
    if (i >= N_NODES * HID) return;
    int n = i >> 6;
    int c = i & (HID - 1);
    atomicAdd(pooled + batch[n] * HID + c, h[i]);
}

// out[g] = pooled[g] . Wf + bf
__global__ void k_head(const float* __restrict__ pooled, const float* __restrict__ Wf,
                       const float* __restrict__ bf, float* __restrict__ out) {
    int g = threadIdx.x;
    if (g >= N_GRAPHS) return;
    float s = bf[0];
    #pragma unroll
    for (int c = 0; c < HID; ++c) s += pooled[g * HID + c] * Wf[c];
    out[g] = s;
}

extern "C" void kernel_launch(void* const* d_in, const int* in_sizes, int n_in,
                              void* d_out, int out_size, void* d_ws, size_t ws_size,
                              hipStream_t stream) {
    (void)in_sizes; (void)n_in; (void)out_size; (void)ws_size;
    const float* x      = (const float*)d_in[0];
    const int*   eidx   = (const int*)d_in[1];
    const float* eattr  = (const float*)d_in[2];
    const int*   batch  = (const int*)d_in[3];
    const float* W[3]   = {(const float*)d_in[4], (const float*)d_in[7],  (const float*)d_in[10]};
    const float* att[3] = {(const float*)d_in[5], (const float*)d_in[8],  (const float*)d_in[11]};
    const float* bia[3] = {(const float*)d_in[6], (const float*)d_in[9],  (const float*)d_in[12]};
    const float* Wf     = (const float*)d_in[13];
    const float* bf     = (const float*)d_in[14];
    const int* src = eidx;               // edge_index[0]
    const int* dst = eidx + N_EDGES;     // edge_index[1]

    // workspace carve-up (~85 MB)
    const size_t NF = (size_t)N_NODES * HID;
    char* w = (char*)d_ws;
    float*    B0     = (float*)w;    w += NF * 4;
    float*    B1     = (float*)w;    w += NF * 4;
    float*    B2     = (float*)w;    w += NF * 4;
    float*    aI     = (float*)w;    w += (size_t)N_NODES * 4;
    float*    aJ     = (float*)w;    w += (size_t)N_NODES * 4;
    float*    segsum = (float*)w;    w += (size_t)N_NODES * 4;
    unsigned* segmax = (unsigned*)w; w += (size_t)N_NODES * 4;
    float*    ebuf   = (float*)w;    w += (size_t)E_TOT * 4;
    float*    pooled = (float*)w;    w += (size_t)N_GRAPHS * HID * 4;

    // buffer rotation: (in, lin, acc, out) per layer; out overwrites lin's buffer
    const float* in_l[3] = {x,  B0, B1};
    float*      lin_l[3] = {B0, B1, B2};
    float*      acc_l[3] = {B1, B2, B0};
    float*      out_l[3] = {B0, B1, B2};

    const int gemm_blocks  = (N_NODES / 16 * 32 + 255) / 256;  // 6250 waves
    const int node_blocks  = (N_NODES + 255) / 256;
    const int nf_blocks    = (int)((NF + 255) / 256);
    const int edge_blocks  = (E_TOT + 255) / 256;
    const int scat_blocks  = (int)(((size_t)E_TOT * 16 + 255) / 256);

    for (int l = 0; l < 3; ++l) {
        k_gemm   <<<gemm_blocks, 256, 0, stream>>>(in_l[l], W[l], lin_l[l]);
        k_alpha  <<<node_blocks, 256, 0, stream>>>(lin_l[l], att[l], aI, aJ, segmax, segsum);
        k_zero   <<<nf_blocks,   256, 0, stream>>>(acc_l[l], (int)NF);
        k_logit  <<<edge_blocks, 256, 0, stream>>>(src, dst, eattr, aI, aJ, att[l], ebuf, segmax);
        k_exp    <<<edge_blocks, 256, 0, stream>>>(dst, segmax, ebuf, segsum);
        k_scatter<<<scat_blocks, 256, 0, stream>>>(src, dst, ebuf, segsum, lin_l[l], acc_l[l]);
        k_final  <<<nf_blocks,   256, 0, stream>>>(acc_l[l], bia[l], out_l[l]);
    }
    k_poolzero<<<16, 256, 0, stream>>>(pooled);
    k_pool    <<<nf_blocks, 256, 0, stream>>>(B2, batch, pooled);
    k_head    <<<1, 64, 0, stream>>>(pooled, Wf, bf, (float*)d_out);
}